// HGNNPBlock_2637109919844
// MI455X (gfx1250) — compile-verified
//
#include <hip/hip_runtime.h>
#include <hip/hip_bf16.h>

typedef __attribute__((ext_vector_type(2))) float v2f;
typedef __attribute__((ext_vector_type(8))) float v8f;

#define BN   8
#define CC   64
#define HID  32
#define LS   3136
#define KK   30
#define BN_SCALE 0.99999500003749969f  /* 1/sqrt(1+1e-5) */
#define FLT_SENTINEL 3.402823466e+38f

// ---------------- sq[i] = sum_c ft[i][c]^2  (over all B*L rows) ----------------
__global__ void sq_kernel(const float* __restrict__ x, float* __restrict__ sq, int total) {
    int i = blockIdx.x * blockDim.x + threadIdx.x;
    if (i >= total) return;
    const float* r = x + (size_t)i * CC;
    float s = 0.f;
    for (int c = 0; c < CC; ++c) { float v = r[c]; s += v * v; }
    sq[i] = s;
}

// ---------------- d2[i][j] = sq[i]+sq[j]-2*(ft@ft^T), V_WMMA_F32_16X16X4_F32 ----------------
// one wave computes a 16x64 output slab: 1 shared A fragment, 4 B fragments,
// 4 INDEPENDENT accumulators -> 4 interleaved WMMA chains (no serial RAW chain)
__global__ __launch_bounds__(32) void dist_kernel(const float* __restrict__ ft,
                                                  const float* __restrict__ sq,
                                                  float* __restrict__ d2) {
    int lane = threadIdx.x;
    int mn   = lane & 15;          // M for A-frag / N for B-frag / N for C
    int kb   = (lane >> 4) << 1;   // K pair base: 0 or 2
    int i0 = blockIdx.x << 4;
    int j0 = blockIdx.y << 6;      // 64-wide slab in j
    const float* arow  = ft + (size_t)(i0 + mn) * CC + kb;
    const float* brow0 = ft + (size_t)(j0 +  0 + mn) * CC + kb;
    const float* brow1 = ft + (size_t)(j0 + 16 + mn) * CC + kb;
    const float* brow2 = ft + (size_t)(j0 + 32 + mn) * CC + kb;
    const float* brow3 = ft + (size_t)(j0 + 48 + mn) * CC + kb;
    v8f c0 = {}, c1 = {}, c2 = {}, c3 = {};
    #pragma unroll
    for (int k0 = 0; k0 < CC; k0 += 4) {
        v2f a  = *(const v2f*)(arow  + k0);
        v2f b0 = *(const v2f*)(brow0 + k0);
        v2f b1 = *(const v2f*)(brow1 + k0);
        v2f b2 = *(const v2f*)(brow2 + k0);
        v2f b3 = *(const v2f*)(brow3 + k0);
        c0 = __builtin_amdgcn_wmma_f32_16x16x4_f32(false, a, false, b0, (short)0, c0, false, false);
        c1 = __builtin_amdgcn_wmma_f32_16x16x4_f32(false, a, false, b1, (short)0, c1, false, false);
        c2 = __builtin_amdgcn_wmma_f32_16x16x4_f32(false, a, false, b2, (short)0, c2, false, false);
        c3 = __builtin_amdgcn_wmma_f32_16x16x4_f32(false, a, false, b3, (short)0, c3, false, false);
    }
    int half = (lane >> 4) << 3;   // 0 or 8
    v8f acc[4] = {c0, c1, c2, c3};
    #pragma unroll
    for (int t = 0; t < 4; ++t) {
        int j = j0 + (t << 4) + mn;
        float sj = sq[j];
        #pragma unroll
        for (int r = 0; r < 8; ++r) {
            int i = i0 + r + half;
            d2[(size_t)i * LS + j] = sq[i] + sj - 2.f * acc[t][r];
        }
    }
}

// ---------------- per-row 30-smallest (ties -> lowest index), LDS-resident ----------------
__global__ void topk_kernel(const float* __restrict__ d2, int* __restrict__ nbr) {
    __shared__ float sv[LS];
    __shared__ float rv[256];
    __shared__ int   ri[256];
    int row = blockIdx.x;
    const float* src = d2 + (size_t)row * LS;
    for (int j = threadIdx.x; j < LS; j += 256) sv[j] = src[j];
    __syncthreads();
    int* out = nbr + (size_t)row * KK;
    for (int it = 0; it < KK; ++it) {
        float best = FLT_SENTINEL; int bi = LS;
        for (int j = threadIdx.x; j < LS; j += 256) {
            float v = sv[j];
            if (v < best) { best = v; bi = j; }
        }
        rv[threadIdx.x] = best; ri[threadIdx.x] = bi;
        __syncthreads();
        for (int s = 128; s > 0; s >>= 1) {
            if (threadIdx.x < s) {
                float v2 = rv[threadIdx.x + s]; int i2 = ri[threadIdx.x + s];
                if (v2 < rv[threadIdx.x] || (v2 == rv[threadIdx.x] && i2 < ri[threadIdx.x])) {
                    rv[threadIdx.x] = v2; ri[threadIdx.x] = i2;
                }
            }
            __syncthreads();
        }
        if (threadIdx.x == 0) { out[it] = ri[0]; sv[ri[0]] = FLT_SENTINEL; }
        __syncthreads();
    }
}

// ---------------- vertex degrees dv[v] = #edges containing v ----------------
__global__ void dv_kernel(const int* __restrict__ nbr, float* __restrict__ dv, int totalE) {
    int e = blockIdx.x * blockDim.x + threadIdx.x;  // global edge b*L + eL
    if (e >= totalE) return;
    int b = e / LS;
    const int* nr = nbr + (size_t)e * KK;
    float* dvb = dv + (size_t)b * LS;
    for (int j = 0; j < KK; ++j) atomicAdd(&dvb[nr[j]], 1.0f);
}

// ---------------- out = bn(X @ W + bias) via WMMA ----------------
// one wave covers all N (NB = N/16 independent accumulators, shared A fragment)
// grid: (L/16, 1, B)
template <int NB>
__global__ __launch_bounds__(32) void gemm_bn_kernel(const float* __restrict__ X,
                                                     const float* __restrict__ W,
                                                     const float* __restrict__ bias,
                                                     const float* __restrict__ g,
                                                     const float* __restrict__ be,
                                                     float* __restrict__ out,
                                                     int Kdim) {
    const int Ndim = NB * 16;
    int lane = threadIdx.x;
    int mn = lane & 15;
    int kb = (lane >> 4) << 1;
    int b  = blockIdx.z;
    int i0 = blockIdx.x << 4;
    const float* Xb = X + (size_t)b * LS * Kdim;
    const float* ar = Xb + (size_t)(i0 + mn) * Kdim + kb;
    v8f c[NB];
    #pragma unroll
    for (int t = 0; t < NB; ++t) c[t] = (v8f){};
    for (int k0 = 0; k0 < Kdim; k0 += 4) {
        v2f a = *(const v2f*)(ar + k0);
        const float* wr0 = W + (size_t)(k0 + kb)     * Ndim + mn;
        const float* wr1 = W + (size_t)(k0 + kb + 1) * Ndim + mn;
        #pragma unroll
        for (int t = 0; t < NB; ++t) {
            v2f bb;
            bb.x = wr0[t << 4];
            bb.y = wr1[t << 4];
            c[t] = __builtin_amdgcn_wmma_f32_16x16x4_f32(false, a, false, bb, (short)0, c[t], false, false);
        }
    }
    float* ob = out + (size_t)b * LS * Ndim;
    int half = (lane >> 4) << 3;
    #pragma unroll
    for (int t = 0; t < NB; ++t) {
        int n = (t << 4) + mn;
        float scale = g[n] * BN_SCALE;
        float bi = bias[n], bo = be[n];
        #pragma unroll
        for (int r = 0; r < 8; ++r) {
            int i = i0 + r + half;
            ob[(size_t)i * Ndim + n] = (c[t][r] + bi) * scale + bo;
        }
    }
}

// ---------------- v2e: Y[e][f] = mean_j X[nbr[e][j]][f] ----------------
__global__ void v2e_kernel(const float* __restrict__ X, const int* __restrict__ nbr,
                           float* __restrict__ Y, int F, int total) {
    int t = blockIdx.x * blockDim.x + threadIdx.x;   // over B*L*F
    if (t >= total) return;
    int f = t % F;
    int e = t / F;          // global edge b*L + eL
    int b = e / LS;
    const int* nr = nbr + (size_t)e * KK;
    const float* Xb = X + (size_t)b * LS * F;
    float s = 0.f;
    for (int j = 0; j < KK; ++j) s += Xb[(size_t)nr[j] * F + f];
    Y[(size_t)e * F + f] = s * (1.0f / KK);
}

// ---------------- e2v scatter: S[nbr[e][j]][f] += Y[e][f] ----------------
__global__ void e2v_kernel(const float* __restrict__ Y, const int* __restrict__ nbr,
                           float* __restrict__ S, int F, int total) {
    int t = blockIdx.x * blockDim.x + threadIdx.x;   // over B*L*F
    if (t >= total) return;
    int f = t % F;
    int e = t / F;
    int b = e / LS;
    float y = Y[(size_t)e * F + f];
    const int* nr = nbr + (size_t)e * KK;
    float* Sb = S + (size_t)b * LS * F;
    for (int j = 0; j < KK; ++j) atomicAdd(&Sb[(size_t)nr[j] * F + f], y);
}

// ---------------- finish: out = S / max(dv,1), optional relu (in-place safe) ----------------
__global__ void finish_kernel(const float* __restrict__ S, const float* __restrict__ dv,
                              float* __restrict__ Out, int F, int relu, int total) {
    int t = blockIdx.x * blockDim.x + threadIdx.x;
    if (t >= total) return;
    int v = t / F;                       // global vertex b*L + vL
    float d = fmaxf(dv[v], 1.0f);
    float val = S[t] / d;
    if (relu) val = fmaxf(val, 0.0f);
    Out[t] = val;
}

__global__ void zero_kernel(float* __restrict__ p, int n) {
    int i = blockIdx.x * blockDim.x + threadIdx.x;
    if (i < n) p[i] = 0.f;
}

static inline int cdiv(int a, int b) { return (a + b - 1) / b; }

extern "C" void kernel_launch(void* const* d_in, const int* in_sizes, int n_in,
                              void* d_out, int out_size, void* d_ws, size_t ws_size,
                              hipStream_t stream) {
    const float* x   = (const float*)d_in[0];
    const float* w1  = (const float*)d_in[1];
    const float* b1  = (const float*)d_in[2];
    const float* g1  = (const float*)d_in[3];
    const float* be1 = (const float*)d_in[4];
    const float* w2  = (const float*)d_in[5];
    const float* b2  = (const float*)d_in[6];
    const float* g2  = (const float*)d_in[7];
    const float* be2 = (const float*)d_in[8];
    float* out = (float*)d_out;

    // workspace layout (bytes, all 256-aligned)
    char* ws = (char*)d_ws;
    const size_t OFF_SQ  = 0;                                  // B*L floats
    const size_t OFF_D2  = OFF_SQ + (size_t)BN * LS * 4;       // L*L floats (39.3 MB)
    const size_t OFF_NBR = OFF_D2 + (size_t)LS * LS * 4;       // B*L*K ints
    const size_t OFF_DV  = OFF_NBR + (size_t)BN * LS * KK * 4; // B*L floats
    float* sq  = (float*)(ws + OFF_SQ);
    float* d2  = (float*)(ws + OFF_D2);
    int*   nbr = (int*)  (ws + OFF_NBR);
    float* dv  = (float*)(ws + OFF_DV);
    // recycled inside the d2 region after top-k is done:
    float* t1 = d2;                              // B*L*32
    float* Y1 = t1 + (size_t)BN * LS * HID;      // B*L*32
    float* s1 = Y1 + (size_t)BN * LS * HID;      // B*L*32
    float* h1 = s1 + (size_t)BN * LS * HID;      // B*L*32
    float* t2 = h1 + (size_t)BN * LS * HID;      // B*L*64
    float* Y2 = t2 + (size_t)BN * LS * CC;       // B*L*64

    const int nV   = BN * LS;        // 25088
    const int nF1  = BN * LS * HID;  // 802816
    const int nF2  = BN * LS * CC;   // 1605632

    // 1) row norms
    sq_kernel<<<cdiv(nV, 256), 256, 0, stream>>>(x, sq, nV);

    // 2) per-sample: fp32 WMMA Gram distances (16x64 slab/wave), then LDS top-k(30)
    for (int b = 0; b < BN; ++b) {
        dist_kernel<<<dim3(LS / 16, LS / 64), 32, 0, stream>>>(
            x + (size_t)b * LS * CC, sq + (size_t)b * LS, d2);
        topk_kernel<<<LS, 256, 0, stream>>>(d2, nbr + (size_t)b * LS * KK);
    }

    // 3) vertex degrees
    zero_kernel<<<cdiv(nV, 256), 256, 0, stream>>>(dv, nV);
    dv_kernel<<<cdiv(nV, 256), 256, 0, stream>>>(nbr, dv, nV);

    // 4) conv1: theta+bn (WMMA), v2e mean, e2v scatter, divide+relu
    gemm_bn_kernel<HID / 16><<<dim3(LS / 16, 1, BN), 32, 0, stream>>>(
        x, w1, b1, g1, be1, t1, CC);
    v2e_kernel<<<cdiv(nF1, 256), 256, 0, stream>>>(t1, nbr, Y1, HID, nF1);
    zero_kernel<<<cdiv(nF1, 256), 256, 0, stream>>>(s1, nF1);
    e2v_kernel<<<cdiv(nF1, 256), 256, 0, stream>>>(Y1, nbr, s1, HID, nF1);
    finish_kernel<<<cdiv(nF1, 256), 256, 0, stream>>>(s1, dv, h1, HID, 1, nF1);

    // 5) conv2: theta+bn (WMMA), v2e mean, e2v scatter into d_out, divide (no relu)
    gemm_bn_kernel<CC / 16><<<dim3(LS / 16, 1, BN), 32, 0, stream>>>(
        h1, w2, b2, g2, be2, t2, HID);
    v2e_kernel<<<cdiv(nF2, 256), 256, 0, stream>>>(t2, nbr, Y2, CC, nF2);
    zero_kernel<<<cdiv(nF2, 256), 256, 0, stream>>>(out, nF2);
    e2v_kernel<<<cdiv(nF2, 256), 256, 0, stream>>>(Y2, nbr, out, CC, nF2);
    finish_kernel<<<cdiv(nF2, 256), 256, 0, stream>>>(out, dv, out, CC, 0, nF2);
}